// PerNeuronSparseReservoir_1245540516176
// MI455X (gfx1250) — compile-verified
//
#include <hip/hip_runtime.h>
#include <stdint.h>

#define B_DIM 256
#define N_DIM 16384
#define DEG   32
#define TN    8            // neurons per block: one per wave32

// ---------------- CDNA5 async global->LDS path (guarded) ----------------
#if defined(__HIP_DEVICE_COMPILE__) && defined(__gfx1250__) && \
    __has_builtin(__builtin_amdgcn_global_load_async_to_lds_b32)
#define USE_ASYNC_LDS 1
#else
#define USE_ASYNC_LDS 0
#endif

#if USE_ASYNC_LDS
typedef __attribute__((address_space(1))) int as1_int;   // global
typedef __attribute__((address_space(3))) int as3_int;   // LDS

__device__ __forceinline__ void async_g2l_b32(const void* g, void* l) {
  // (global src AS1 int*, lds dst AS3 int*, imm offset, imm cpol)
  __builtin_amdgcn_global_load_async_to_lds_b32(
      (as1_int*)g, (as3_int*)l, 0, 0);
}
__device__ __forceinline__ void wait_async0() {
#if __has_builtin(__builtin_amdgcn_s_wait_asynccnt)
  __builtin_amdgcn_s_wait_asynccnt(0);
#else
  asm volatile("s_wait_asynccnt 0x0" ::: "memory");
#endif
}
#endif

// ---------------- Kernel 1: x (B x N) -> xT (N x B), LDS-tiled ----------------
// Both phases fully coalesced (128B per wave); HBM-bound pre-pass.
__global__ __launch_bounds__(256) void pnsr_transpose_kernel(
    const float* __restrict__ x, float* __restrict__ xT) {
  __shared__ float tile[32][33];           // +1 pad: conflict-free transpose
  const int n0 = blockIdx.x * 32;
  const int b0 = blockIdx.y * 32;
  const int tx = threadIdx.x;              // 0..31
  const int ty = threadIdx.y;              // 0..7
#pragma unroll
  for (int i = 0; i < 32; i += 8)
    tile[ty + i][tx] = x[(size_t)(b0 + ty + i) * N_DIM + (n0 + tx)];
  __syncthreads();
#pragma unroll
  for (int i = 0; i < 32; i += 8)
    xT[(size_t)(n0 + ty + i) * B_DIM + (b0 + tx)] = tile[tx][ty + i];
}

// ---------------- Kernel 2: out = ReLU(x * M)  (column-AXPY SpMM) -------------
// Compile-time strides => immediate-offset global addressing.
// Fast path <SB=1, SC=B_DIM>: xsrc = xT, b128 column loads.
// Fallback  <SB=N_DIM, SC=1>: xsrc = x (no workspace), scalar loads.
template <int SB, int SC>
__global__ __launch_bounds__(256) void pnsr_spmm_kernel(
    const float* __restrict__ xsrc, const float* __restrict__ w,
    const int* __restrict__ idx, float* __restrict__ out) {
  __shared__ int2  s_pair[TN * DEG];                // {idx, w bits}: 2 KB
  __shared__ float s_out[B_DIM * (TN + 1)];         // 9 KB, stride 9 -> bank-free

  const int t  = threadIdx.x;                       // 0..255 == TN*DEG slots
  const int n0 = blockIdx.x * TN;

#if USE_ASYNC_LDS
  async_g2l_b32(idx + (size_t)n0 * DEG + t, &s_pair[t].x);
  async_g2l_b32(w   + (size_t)n0 * DEG + t, &s_pair[t].y);
  wait_async0();
#else
  s_pair[t].x = idx[(size_t)n0 * DEG + t];
  s_pair[t].y = __float_as_int(w[(size_t)n0 * DEG + t]);
#endif
  __syncthreads();

  const int lane = t & 31;                          // wave32
  const int j    = t >> 5;                          // wave id == local neuron

  if constexpr (SB == 1) {
    // -------- fast path: 2 x b128 coalesced loads per nonzero --------
    float4 acc0 = {0.f, 0.f, 0.f, 0.f};
    float4 acc1 = {0.f, 0.f, 0.f, 0.f};
#pragma unroll 8
    for (int d = 0; d < DEG; ++d) {
      const int2  p  = s_pair[j * DEG + d];         // one ds_load_b64 broadcast
      const float wv = __int_as_float(p.y);
      const float4* col4 = (const float4*)(xsrc + (size_t)p.x * SC);
      const float4 a = col4[lane];                  // b = lane*4+q
      const float4 b = col4[lane + 32];             // b = 128+lane*4+q
      acc0.x += wv * a.x; acc0.y += wv * a.y;
      acc0.z += wv * a.z; acc0.w += wv * a.w;
      acc1.x += wv * b.x; acc1.y += wv * b.y;
      acc1.z += wv * b.z; acc1.w += wv * b.w;
    }
    const float* a0 = (const float*)&acc0;
    const float* a1 = (const float*)&acc1;
#pragma unroll
    for (int q = 0; q < 4; ++q) {
      s_out[(lane * 4 + q) * (TN + 1) + j]         = a0[q];
      s_out[(128 + lane * 4 + q) * (TN + 1) + j]   = a1[q];
    }
  } else {
    // -------- fallback: direct gather from row-major x --------
    float acc[8] = {0.f, 0.f, 0.f, 0.f, 0.f, 0.f, 0.f, 0.f};
#pragma unroll 4
    for (int d = 0; d < DEG; ++d) {
      const int2  p  = s_pair[j * DEG + d];
      const float wv = __int_as_float(p.y);
      const float* col = xsrc + (size_t)p.x * SC;
#pragma unroll
      for (int i = 0; i < 8; ++i)                   // b = i*32+lane
        acc[i] += wv * col[(size_t)(i * 32 + lane) * SB];
    }
#pragma unroll
    for (int i = 0; i < 8; ++i)
      s_out[(i * 32 + lane) * (TN + 1) + j] = acc[i];
  }
  __syncthreads();

  // thread t == batch row b: ReLU + two coalesced float4 stores (32B-aligned)
  float r[TN];
#pragma unroll
  for (int k = 0; k < TN; ++k) {
    const float v = s_out[t * (TN + 1) + k];
    r[k] = v > 0.f ? v : 0.f;
  }
  float* op = out + (size_t)t * N_DIM + n0;
  *(float4*)(op + 0) = make_float4(r[0], r[1], r[2], r[3]);
  *(float4*)(op + 4) = make_float4(r[4], r[5], r[6], r[7]);
}

// ---------------- Host launcher ----------------
extern "C" void kernel_launch(void* const* d_in, const int* in_sizes, int n_in,
                              void* d_out, int out_size, void* d_ws, size_t ws_size,
                              hipStream_t stream) {
  const float* x   = (const float*)d_in[0];   // (B, N) f32
  const float* w   = (const float*)d_in[1];   // (N, DEG) f32
  const int*   idx = (const int*)d_in[2];     // (N, DEG) i32
  float*       out = (float*)d_out;           // (B, N) f32

  const size_t xT_bytes = (size_t)N_DIM * B_DIM * sizeof(float);
  if (d_ws != nullptr && ws_size >= xT_bytes) {
    float* xT = (float*)d_ws;
    dim3 tb(32, 8);
    dim3 tg(N_DIM / 32, B_DIM / 32);
    pnsr_transpose_kernel<<<tg, tb, 0, stream>>>(x, xT);
    pnsr_spmm_kernel<1, B_DIM><<<N_DIM / TN, 256, 0, stream>>>(xT, w, idx, out);
  } else {
    // workspace too small: direct (uncoalesced) path, still correct
    pnsr_spmm_kernel<N_DIM, 1><<<N_DIM / TN, 256, 0, stream>>>(x, w, idx, out);
  }
}